// RoutingLayers_71279277244617
// MI455X (gfx1250) — compile-verified
//
#include <hip/hip_runtime.h>
#include <math.h>

// Problem geometry (fixed by the reference, but N/E derived from in_sizes).
#define CAPS 4
#define DIM  32
#define ROW  (CAPS * DIM)        // 128 floats = 512 bytes per node row
#define EPS_CLAMP 1e-8f
#define EPS_NORM  1e-12f

// Native CDNA5 tanh (V_TANH_F32) when the toolchain exposes it; libm otherwise.
#if defined(__has_builtin)
#if __has_builtin(__builtin_amdgcn_tanhf)
#define FAST_TANH(x) __builtin_amdgcn_tanhf(x)
#endif
#endif
#ifndef FAST_TANH
#define FAST_TANH(x) tanhf(x)
#endif

// ---------------------------------------------------------------------------
// Pass 1: per-edge softmax over the 4 capsule logits, store norm, and
// scatter-add into deg[src] with f32 L2 atomics.  One thread per edge,
// float4 (b128) load/store of the 4-wide rows.
// ---------------------------------------------------------------------------
__global__ void __launch_bounds__(256)
k_softmax_deg(const float4* __restrict__ edge_ncaps,
              const int*    __restrict__ src_trg,
              float4*       __restrict__ norm_out,
              float*        __restrict__ deg,
              int E) {
    int e = blockIdx.x * blockDim.x + threadIdx.x;
    if (e >= E) return;
    float4 v = edge_ncaps[e];
    float m = fmaxf(fmaxf(v.x, v.y), fmaxf(v.z, v.w));
    float a = __expf(v.x - m);
    float b = __expf(v.y - m);
    float c = __expf(v.z - m);
    float d = __expf(v.w - m);
    float r = 1.0f / (a + b + c + d);
    float4 n;
    n.x = a * r; n.y = b * r; n.z = c * r; n.w = d * r;
    norm_out[e] = n;

    int s = src_trg[e];
    float* dp = deg + (size_t)s * CAPS;
    atomicAdd(dp + 0, n.x);
    atomicAdd(dp + 1, n.y);
    atomicAdd(dp + 2, n.z);
    atomicAdd(dp + 3, n.w);
}

// ---------------------------------------------------------------------------
// Pass 2: deg -> 1/sqrt(max(deg, 1e-8)) in place. Elementwise over N*CAPS.
// ---------------------------------------------------------------------------
__global__ void __launch_bounds__(256)
k_inv_sqrt(float* __restrict__ deg, int n) {
    int i = blockIdx.x * blockDim.x + threadIdx.x;
    if (i >= n) return;
    float d = fmaxf(deg[i], EPS_CLAMP);
    deg[i] = 1.0f / sqrtf(d);
}

// ---------------------------------------------------------------------------
// Pass 3: wave-per-edge weighted feature scatter.
//   ncap[src] += x[trg] * (norm * inv[src] * inv[trg]) per capsule.
// Lane l of the wave32 handles float4 chunk l of the 128-float row
// (cap = l>>3), so the 512B row load is one fully-coalesced burst and the
// 4 atomics/lane land on 4 consecutive L2 cachelines.
// ---------------------------------------------------------------------------
__global__ void __launch_bounds__(256)
k_scatter(const float4* __restrict__ x,      // N rows x 32 float4
          const float*  __restrict__ norm,   // [E, 4]
          const float*  __restrict__ inv,    // [N, 4]
          const int*    __restrict__ src_trg,
          const int*    __restrict__ trg_src,
          float*        __restrict__ ncap,   // [N, 128] (d_out)
          int E) {
    int t    = blockIdx.x * blockDim.x + threadIdx.x;
    int e    = t >> 5;
    int lane = t & 31;
    if (e >= E) return;

    int src = src_trg[e];
    int trg = trg_src[e];
    int cap = lane >> 3;

    float w = norm[(size_t)e * CAPS + cap]
            * inv[(size_t)src * CAPS + cap]
            * inv[(size_t)trg * CAPS + cap];

    float4 v = x[(size_t)trg * (ROW / 4) + lane];

    float* op = ncap + (size_t)src * ROW + lane * 4;
    atomicAdd(op + 0, v.x * w);
    atomicAdd(op + 1, v.y * w);
    atomicAdd(op + 2, v.z * w);
    atomicAdd(op + 3, v.w * w);
}

// ---------------------------------------------------------------------------
// Pass 4: wave-per-edge capsule scores.
//   new_edge_ncaps[e,c] = < l2norm(ncap[src,c,:]) , tanh(l2norm(x[trg,c,:])) >
// 8 lanes per capsule (4 caps * 8 lanes = wave32); each lane owns a float4 of
// the 32-float capsule vector. Sub-wave reductions via __shfl_xor (lane
// swizzle hardware, no LDS memory, no barriers). tanh uses the native CDNA5
// V_TANH_F32 transcendental when available — this is the FLOP-heavy pass
// (E*C*D = 64M tanh evaluations).
// ---------------------------------------------------------------------------
__global__ void __launch_bounds__(256)
k_scores(const float4* __restrict__ x,      // N rows x 32 float4
         const float4* __restrict__ ncap,   // N rows x 32 float4
         const int*    __restrict__ src_trg,
         const int*    __restrict__ trg_src,
         float*        __restrict__ out_edge, // [E, 4]
         int E) {
    int t    = blockIdx.x * blockDim.x + threadIdx.x;
    int e    = t >> 5;
    int lane = t & 31;
    if (e >= E) return;

    int src = src_trg[e];
    int trg = trg_src[e];
    int cap = lane >> 3;    // capsule handled by this 8-lane group
    int g   = lane & 7;     // position within the group

    size_t base = (size_t)cap * (DIM / 4) + g;
    float4 h  = ncap[(size_t)src * (ROW / 4) + base];
    float4 tl = x   [(size_t)trg * (ROW / 4) + base];

    float ssh = h.x * h.x + h.y * h.y + h.z * h.z + h.w * h.w;
    float sst = tl.x * tl.x + tl.y * tl.y + tl.z * tl.z + tl.w * tl.w;
    #pragma unroll
    for (int m = 1; m < 8; m <<= 1) {
        ssh += __shfl_xor(ssh, m, 32);
        sst += __shfl_xor(sst, m, 32);
    }
    float hi = 1.0f / fmaxf(sqrtf(ssh), EPS_NORM);
    float ti = 1.0f / fmaxf(sqrtf(sst), EPS_NORM);

    float dot = h.x * hi * FAST_TANH(tl.x * ti)
              + h.y * hi * FAST_TANH(tl.y * ti)
              + h.z * hi * FAST_TANH(tl.z * ti)
              + h.w * hi * FAST_TANH(tl.w * ti);
    #pragma unroll
    for (int m = 1; m < 8; m <<= 1) {
        dot += __shfl_xor(dot, m, 32);
    }
    if (g == 0) {
        out_edge[(size_t)e * CAPS + cap] = dot;
    }
}

// ---------------------------------------------------------------------------
// Host-side launch (graph-capture safe: only kernels + hipMemsetAsync).
// ---------------------------------------------------------------------------
extern "C" void kernel_launch(void* const* d_in, const int* in_sizes, int n_in,
                              void* d_out, int out_size, void* d_ws, size_t ws_size,
                              hipStream_t stream) {
    // Input order: x, edge_ncaps, iid2tid (unused), tid2iid (unused),
    //              src_trg, trg_src
    const float* x          = (const float*)d_in[0];
    const float* edge_ncaps = (const float*)d_in[1];
    const int*   src_trg    = (const int*)  d_in[4];
    const int*   trg_src    = (const int*)  d_in[5];

    const int N = in_sizes[0] / ROW;   // 100000
    const int E = in_sizes[4];         // 500000

    // Workspace: deg/inv [N*4] floats, then norm [E*4] floats (~9.6 MB).
    float* deg  = (float*)d_ws;
    float* norm = deg + (size_t)N * CAPS;

    // Output layout: new_x [N*128] floats, then new_edge_ncaps [E*4] floats.
    float* out_x    = (float*)d_out;
    float* out_edge = out_x + (size_t)N * ROW;

    // Zero the accumulation targets every call (atomics accumulate into them;
    // the harness does not re-poison between timed replays).
    hipMemsetAsync(deg,   0, (size_t)N * CAPS * sizeof(float), stream);
    hipMemsetAsync(out_x, 0, (size_t)N * ROW  * sizeof(float), stream);

    const int TPB = 256;

    // Pass 1: softmax + degree scatter (thread per edge).
    k_softmax_deg<<<(E + TPB - 1) / TPB, TPB, 0, stream>>>(
        (const float4*)edge_ncaps, src_trg, (float4*)norm, deg, E);

    // Pass 2: deg -> 1/sqrt(clamp(deg)).
    int nc = N * CAPS;
    k_inv_sqrt<<<(nc + TPB - 1) / TPB, TPB, 0, stream>>>(deg, nc);

    // Pass 3: weighted feature scatter (wave32 per edge).
    long long threads = (long long)E * 32;
    int blocks = (int)((threads + TPB - 1) / TPB);
    k_scatter<<<blocks, TPB, 0, stream>>>(
        (const float4*)x, norm, deg, src_trg, trg_src, out_x, E);

    // Pass 4: per-edge capsule scores (wave32 per edge).
    k_scores<<<blocks, TPB, 0, stream>>>(
        (const float4*)x, (const float4*)out_x, src_trg, trg_src, out_edge, E);
}